// GATClassifier_3504693313922
// MI455X (gfx1250) — compile-verified
//
#include <hip/hip_runtime.h>

typedef __attribute__((ext_vector_type(2))) float v2f;
typedef __attribute__((ext_vector_type(8))) float v8f;

#define N_NODES  50000
#define N_EDGES  800000
#define E_TOT    (N_EDGES + N_NODES)   /* 850000 incl. self loops */
#define IN_DIM   128
#define HID      64
#define HEADS    2
#define N_GRAPHS 512
#define NEG_SLOPE 0.2f

static inline int cdiv(long long a, long long b) { return (int)((a + b - 1) / b); }

// ---------------------------------------------------------------------------
// FP32 GEMM via V_WMMA_F32_16X16X4_F32. One wave per 16x16 C tile.
// A: [M,K] row-major, B: [K,N] row-major, C: [M,N].
// A-frag layout (16x4 fp32): lane&15 = M row, (lane>>4)*2 + j = K (j = vgpr).
// B-frag layout (4x16 fp32): lane&15 = N col, (lane>>4)*2 + j = K.
// C/D layout (16x16 fp32):   lane&15 = N col, M = vgpr_i + 8*(lane>>4).
// ---------------------------------------------------------------------------
__global__ void gemm_wmma_f32(const float* __restrict__ A, const float* __restrict__ B,
                              float* __restrict__ C, int M, int K, int N) {
  const int wave   = blockIdx.x * (blockDim.x >> 5) + (threadIdx.x >> 5);
  const int nTiles = N >> 4;
  const int mT     = wave / nTiles;
  const int nT     = wave - mT * nTiles;
  if ((mT << 4) >= M) return;                    // uniform per wave
  const int lane = threadIdx.x & 31;
  const int r    = lane & 15;
  const int g    = lane >> 4;

  const float* Arow = A + (size_t)((mT << 4) + r) * K;
  const float* Bcol = B + (nT << 4) + r;

  v8f acc = {};
  for (int k = 0; k < K; k += 4) {
    const int ka = k + (g << 1);
    v2f a = *(const v2f*)(Arow + ka);            // 8B-aligned (ka even)
    v2f b;
    b.x = Bcol[(size_t)(ka + 0) * N];
    b.y = Bcol[(size_t)(ka + 1) * N];
    acc = __builtin_amdgcn_wmma_f32_16x16x4_f32(
        /*neg_a=*/false, a, /*neg_b=*/false, b,
        /*c_mod=*/(short)0, acc, /*reuse_a=*/false, /*reuse_b=*/false);
  }

  float* Crow = C + (size_t)((mT << 4) + (g << 3)) * N + (nT << 4) + r;
#pragma unroll
  for (int i = 0; i < 8; ++i) Crow[(size_t)i * N] = acc[i];
}

// ---------------------------------------------------------------------------
// alpha_s[n,h] = sum_c h[n,h,c]*a_src[h,c]  (same for alpha_d). One wave per
// (node, head); HID=64 -> 2 elements per lane, wave32 xor-shuffle reduce.
// ---------------------------------------------------------------------------
__global__ void alpha_kernel(const float* __restrict__ h, const float* __restrict__ aS,
                             const float* __restrict__ aD, float* __restrict__ oS,
                             float* __restrict__ oD, int N, int H) {
  const int wave = blockIdx.x * (blockDim.x >> 5) + (threadIdx.x >> 5);
  if (wave >= N * H) return;
  const int n    = wave / H;
  const int head = wave - n * H;
  const int lane = threadIdx.x & 31;
  const float* hp  = h + (size_t)n * H * HID + (size_t)head * HID;
  const float* asp = aS + head * HID;
  const float* adp = aD + head * HID;
  const float h0 = hp[lane], h1 = hp[lane + 32];
  float ps = h0 * asp[lane] + h1 * asp[lane + 32];
  float pd = h0 * adp[lane] + h1 * adp[lane + 32];
#pragma unroll
  for (int m = 16; m >= 1; m >>= 1) {
    ps += __shfl_xor(ps, m, 32);
    pd += __shfl_xor(pd, m, 32);
  }
  if (lane == 0) { oS[wave] = ps; oD[wave] = pd; }
}

__global__ void fill_f32(float* __restrict__ p, float v, int n) {
  int i = blockIdx.x * blockDim.x + threadIdx.x;
  if (i < n) p[i] = v;
}

// e = leaky_relu(alpha_s[src]+alpha_d[dst]); store; segment-max into emax[dst]
__global__ void edge_scores(const float* __restrict__ aS, const float* __restrict__ aD,
                            const int* __restrict__ ei, int H,
                            float* __restrict__ ebuf, float* __restrict__ emax) {
  int tid = blockIdx.x * blockDim.x + threadIdx.x;
  if (tid >= E_TOT * H) return;
  int e = tid / H, hd = tid - e * H;
  int s, d;
  if (e < N_EDGES) { s = ei[e]; d = ei[N_EDGES + e]; }
  else             { s = d = e - N_EDGES; }
  float v = aS[s * H + hd] + aD[d * H + hd];
  v = v > 0.f ? v : NEG_SLOPE * v;
  ebuf[tid] = v;
  atomicMax(&emax[d * H + hd], v);
}

// ex = exp(e - emax[dst]); store; segment-sum into denom[dst]
__global__ void edge_exp(const int* __restrict__ ei, int H, float* __restrict__ ebuf,
                         const float* __restrict__ emax, float* __restrict__ den) {
  int tid = blockIdx.x * blockDim.x + threadIdx.x;
  if (tid >= E_TOT * H) return;
  int e = tid / H, hd = tid - e * H;
  int d = (e < N_EDGES) ? ei[N_EDGES + e] : (e - N_EDGES);
  float ex = expf(ebuf[tid] - emax[d * H + hd]);
  ebuf[tid] = ex;
  atomicAdd(&den[d * H + hd], ex);
}

// coef = ex / denom[dst]
__global__ void edge_norm(const int* __restrict__ ei, int H, float* __restrict__ ebuf,
                          const float* __restrict__ den) {
  int tid = blockIdx.x * blockDim.x + threadIdx.x;
  if (tid >= E_TOT * H) return;
  int e = tid / H, hd = tid - e * H;
  int d = (e < N_EDGES) ? ei[N_EDGES + e] : (e - N_EDGES);
  ebuf[tid] = ebuf[tid] / den[d * H + hd];
}

// agg[dst, ch] += h[src, ch] * coef[e, ch>>6]; one thread per (edge, channel).
// CH = H*64 (power of 2). Coalesced 512B/256B gather of h[src].
__global__ void edge_agg(const float* __restrict__ h, const float* __restrict__ coef,
                         const int* __restrict__ ei, int H, int chShift,
                         float* __restrict__ agg) {
  unsigned tid = blockIdx.x * blockDim.x + threadIdx.x;
  const unsigned CH = (unsigned)H << 6;
  if (tid >= (unsigned)E_TOT * CH) return;
  unsigned e  = tid >> chShift;
  unsigned ch = tid & (CH - 1);
  int hd = (int)(ch >> 6);
  int s, d;
  if (e < N_EDGES) { s = ei[e]; d = ei[N_EDGES + e]; }
  else             { s = d = (int)e - N_EDGES; }
  float c = coef[(size_t)e * H + hd];
  atomicAdd(&agg[(size_t)d * CH + ch], h[(size_t)s * CH + ch] * c);
}

__global__ void bias_relu(float* __restrict__ x, const float* __restrict__ b,
                          int n, int cMask) {
  int i = blockIdx.x * blockDim.x + threadIdx.x;
  if (i >= n) return;
  float v = x[i] + b[i & cMask];
  x[i] = v > 0.f ? v : 0.f;
}

__global__ void pool_kernel(const float* __restrict__ h, const int* __restrict__ batch,
                            float* __restrict__ pooled, float* __restrict__ cnt) {
  int tid = blockIdx.x * blockDim.x + threadIdx.x;
  if (tid >= N_NODES * HID) return;
  int n = tid >> 6, c = tid & 63;
  int g = batch[n];
  atomicAdd(&pooled[g * HID + c], h[tid]);
  if (c == 0) atomicAdd(&cnt[g], 1.0f);
}

__global__ void head_kernel(const float* __restrict__ pooled, const float* __restrict__ cnt,
                            const float* __restrict__ fcW, const float* __restrict__ fcb,
                            float* __restrict__ out) {
  int g = blockIdx.x * blockDim.x + threadIdx.x;
  if (g >= N_GRAPHS) return;
  float c = cnt[g]; if (c < 1.f) c = 1.f;
  float inv = 1.f / c;
  float l0 = fcb[0], l1 = fcb[1];
#pragma unroll 8
  for (int i = 0; i < HID; ++i) {
    float p = pooled[(size_t)g * HID + i] * inv;
    l0 += p * fcW[i * 2 + 0];
    l1 += p * fcW[i * 2 + 1];
  }
  float m   = fmaxf(l0, l1);
  float lse = m + logf(expf(l0 - m) + expf(l1 - m));
  out[g * 2 + 0] = l0 - lse;
  out[g * 2 + 1] = l1 - lse;
}

// ---------------------------------------------------------------------------
extern "C" void kernel_launch(void* const* d_in, const int* in_sizes, int n_in,
                              void* d_out, int out_size, void* d_ws, size_t ws_size,
                              hipStream_t stream) {
  (void)in_sizes; (void)n_in; (void)out_size; (void)ws_size;
  const float* x     = (const float*)d_in[0];
  const int*   ei    = (const int*)d_in[1];    // [2, N_EDGES] (row0=src, row1=dst)
  const int*   batch = (const int*)d_in[2];
  const float* W1    = (const float*)d_in[3];
  const float* aS1   = (const float*)d_in[4];
  const float* aD1   = (const float*)d_in[5];
  const float* b1    = (const float*)d_in[6];
  const float* W2    = (const float*)d_in[7];
  const float* aS2   = (const float*)d_in[8];
  const float* aD2   = (const float*)d_in[9];
  const float* b2    = (const float*)d_in[10];
  const float* fcW   = (const float*)d_in[11];
  const float* fcb   = (const float*)d_in[12];
  float* out = (float*)d_out;

  char*  ws  = (char*)d_ws;
  size_t off = 0;
  auto alloc = [&](size_t elems) {
    float* p = (float*)(ws + off);
    off += ((elems * sizeof(float)) + 255) & ~(size_t)255;
    return p;
  };
  float* h1     = alloc((size_t)N_NODES * 128);   // layer1 GEMM output
  float* x2     = alloc((size_t)N_NODES * 128);   // layer1 aggregate -> relu -> layer2 input
  float* h2     = alloc((size_t)N_NODES * 64);    // layer2 GEMM output
  float* hout   = alloc((size_t)N_NODES * 64);    // layer2 aggregate -> relu
  float* ebuf   = alloc((size_t)E_TOT * HEADS);   // edge scores / exp / coef
  float* alS    = alloc((size_t)N_NODES * HEADS);
  float* alD    = alloc((size_t)N_NODES * HEADS);
  float* emax   = alloc((size_t)N_NODES * HEADS);
  float* den    = alloc((size_t)N_NODES * HEADS);
  float* pooled = alloc((size_t)N_GRAPHS * HID);
  float* cnt    = alloc((size_t)N_GRAPHS);

  const int B = 256;

  // ----------------- Layer 1 (H=2, C=64) -----------------
  gemm_wmma_f32<<<cdiv((N_NODES / 16) * (128 / 16), 8), B, 0, stream>>>(
      x, W1, h1, N_NODES, 128, 128);
  alpha_kernel<<<cdiv((long long)N_NODES * HEADS * 32, B), B, 0, stream>>>(
      h1, aS1, aD1, alS, alD, N_NODES, HEADS);

  fill_f32<<<cdiv(N_NODES * HEADS, B), B, 0, stream>>>(emax, -3.0e38f, N_NODES * HEADS);
  fill_f32<<<cdiv(N_NODES * HEADS, B), B, 0, stream>>>(den, 0.f, N_NODES * HEADS);
  fill_f32<<<cdiv((long long)N_NODES * 128, B), B, 0, stream>>>(x2, 0.f, N_NODES * 128);

  edge_scores<<<cdiv((long long)E_TOT * HEADS, B), B, 0, stream>>>(alS, alD, ei, HEADS, ebuf, emax);
  edge_exp<<<cdiv((long long)E_TOT * HEADS, B), B, 0, stream>>>(ei, HEADS, ebuf, emax, den);
  edge_norm<<<cdiv((long long)E_TOT * HEADS, B), B, 0, stream>>>(ei, HEADS, ebuf, den);
  edge_agg<<<cdiv((long long)E_TOT * 128, B), B, 0, stream>>>(h1, ebuf, ei, HEADS, 7, x2);
  bias_relu<<<cdiv((long long)N_NODES * 128, B), B, 0, stream>>>(x2, b1, N_NODES * 128, 127);

  // ----------------- Layer 2 (H=1, C=64) -----------------
  gemm_wmma_f32<<<cdiv((N_NODES / 16) * (64 / 16), 8), B, 0, stream>>>(
      x2, W2, h2, N_NODES, 128, 64);
  alpha_kernel<<<cdiv((long long)N_NODES * 32, B), B, 0, stream>>>(
      h2, aS2, aD2, alS, alD, N_NODES, 1);

  fill_f32<<<cdiv(N_NODES, B), B, 0, stream>>>(emax, -3.0e38f, N_NODES);
  fill_f32<<<cdiv(N_NODES, B), B, 0, stream>>>(den, 0.f, N_NODES);
  fill_f32<<<cdiv((long long)N_NODES * 64, B), B, 0, stream>>>(hout, 0.f, N_NODES * 64);

  edge_scores<<<cdiv(E_TOT, B), B, 0, stream>>>(alS, alD, ei, 1, ebuf, emax);
  edge_exp<<<cdiv(E_TOT, B), B, 0, stream>>>(ei, 1, ebuf, emax, den);
  edge_norm<<<cdiv(E_TOT, B), B, 0, stream>>>(ei, 1, ebuf, den);
  edge_agg<<<cdiv((long long)E_TOT * 64, B), B, 0, stream>>>(h2, ebuf, ei, 1, 6, hout);
  bias_relu<<<cdiv((long long)N_NODES * 64, B), B, 0, stream>>>(hout, b2, N_NODES * 64, 63);

  // ----------------- Pool + FC + log_softmax -----------------
  fill_f32<<<cdiv(N_GRAPHS * HID, B), B, 0, stream>>>(pooled, 0.f, N_GRAPHS * HID);
  fill_f32<<<cdiv(N_GRAPHS, B), B, 0, stream>>>(cnt, 0.f, N_GRAPHS);
  pool_kernel<<<cdiv((long long)N_NODES * HID, B), B, 0, stream>>>(hout, batch, pooled, cnt);
  head_kernel<<<cdiv(N_GRAPHS, B), B, 0, stream>>>(pooled, cnt, fcW, fcb, out);
}